// TrueGenerativeModel_34170759807476
// MI455X (gfx1250) — compile-verified
//
#include <hip/hip_runtime.h>
#include <hip/hip_bf16.h>

// Shapes from the reference:
//   program_id      : (128, 64)            int32          -> 8192 programs
//   hearts_obs      : (128, 64, 2, 64, 64) float32        -> per program: 2 planes of 4096 floats
//   rectangles_obs  : (128, 64, 2, 64, 64) float32
//   out             : (128, 64, 64, 64)    float32        -> per program: 4096 floats
//
// out[p, i] = clip(c0*h1[i] + c1*h2[i] + c2*r1[i] + c3*r2[i], 0, 1)
// where c = COEFFS[program_id[p]].
//
// Pure HBM-bandwidth-bound (~640 MB traffic, ~0.27 GFLOP). Strategy:
//  - block-per-program so program_id / coefficients are wave-uniform (SGPRs)
//  - float4 (B128) non-temporal loads/stores, fully unrolled so 16 loads are
//    in flight per thread before the first s_wait_loadcnt
//  - NT hints keep the 512 MB one-shot input stream from thrashing L2 (192 MB)

typedef float v4f __attribute__((ext_vector_type(4)));

#define PLANE 4096        // 64*64 floats per plane
#define OBS_STRIDE 8192   // 2 planes per program per obs tensor
#define VEC_PER_PLANE (PLANE / 4)   // 1024 float4 per plane
#define THREADS 256
#define ITERS (VEC_PER_PLANE / THREADS)  // 4

__constant__ float COEFFS_c[9][4] = {
    { 1.0f,  0.0f,  0.0f,  0.0f},
    { 0.0f,  0.0f,  1.0f,  0.0f},
    { 1.0f,  0.0f,  1.0f,  0.0f},
    { 1.0f,  1.0f,  0.0f,  0.0f},
    { 0.0f,  0.0f,  1.0f,  1.0f},
    { 1.0f, -1.0f,  0.0f,  0.0f},
    { 1.0f,  0.0f, -1.0f,  0.0f},
    {-1.0f,  0.0f,  1.0f,  0.0f},
    { 0.0f,  0.0f,  1.0f, -1.0f}
};

__global__ void __launch_bounds__(THREADS)
true_gen_model_kernel(const int* __restrict__ prog,
                      const float* __restrict__ hearts,
                      const float* __restrict__ rects,
                      float* __restrict__ out)
{
    const int p = blockIdx.x;

    // Block-uniform: compiler scalarizes these into s_load + SGPR broadcasts.
    const int pid = prog[p];
    const float c0 = COEFFS_c[pid][0];
    const float c1 = COEFFS_c[pid][1];
    const float c2 = COEFFS_c[pid][2];
    const float c3 = COEFFS_c[pid][3];

    const v4f* __restrict__ h1 =
        (const v4f*)(hearts + (size_t)p * OBS_STRIDE);
    const v4f* __restrict__ h2 = h1 + VEC_PER_PLANE;
    const v4f* __restrict__ r1 =
        (const v4f*)(rects + (size_t)p * OBS_STRIDE);
    const v4f* __restrict__ r2 = r1 + VEC_PER_PLANE;
    v4f* __restrict__ o = (v4f*)(out + (size_t)p * PLANE);

    const int t = threadIdx.x;

    // Issue all 16 B128 loads before any use: maximizes outstanding
    // LOADcnt so the wave covers HBM latency on this streaming kernel.
    v4f va[ITERS], vb[ITERS], vc[ITERS], vd[ITERS];
#pragma unroll
    for (int j = 0; j < ITERS; ++j) {
        const int i = t + j * THREADS;
        va[j] = __builtin_nontemporal_load(&h1[i]);
        vb[j] = __builtin_nontemporal_load(&h2[i]);
        vc[j] = __builtin_nontemporal_load(&r1[i]);
        vd[j] = __builtin_nontemporal_load(&r2[i]);
    }

#pragma unroll
    for (int j = 0; j < ITERS; ++j) {
        const int i = t + j * THREADS;
        v4f acc;
#pragma unroll
        for (int k = 0; k < 4; ++k) {
            float a = c0 * va[j][k];
            a = fmaf(c1, vb[j][k], a);
            a = fmaf(c2, vc[j][k], a);
            a = fmaf(c3, vd[j][k], a);
            acc[k] = fminf(fmaxf(a, 0.0f), 1.0f);
        }
        __builtin_nontemporal_store(acc, &o[i]);
    }
}

extern "C" void kernel_launch(void* const* d_in, const int* in_sizes, int n_in,
                              void* d_out, int out_size, void* d_ws, size_t ws_size,
                              hipStream_t stream) {
    const int*   prog   = (const int*)d_in[0];
    const float* hearts = (const float*)d_in[1];
    const float* rects  = (const float*)d_in[2];
    float*       out    = (float*)d_out;

    const int nprog = in_sizes[0];   // 128*64 = 8192 programs
    true_gen_model_kernel<<<nprog, THREADS, 0, stream>>>(prog, hearts, rects, out);
}